// InvariantMaxLayer_74053826118101
// MI455X (gfx1250) — compile-verified
//
#include <hip/hip_runtime.h>
#include <hip/hip_bf16.h>

typedef float v4f __attribute__((ext_vector_type(4)));
typedef float v2f __attribute__((ext_vector_type(2)));
typedef float v8f __attribute__((ext_vector_type(8)));

// x: (16, 512, 512, 64) fp32, flat 268,435,456 floats = 2048 slabs of 131072 floats.
// Pass 1: one block per slab (4 contiguous (b,i) rows). Each thread accumulates a
// float4 whose channel group is fixed: float4 index tid + 256k -> float index
// 4*tid + 1024k, channel = (4*tid) % 64 for all k. Block-level cross-thread
// reduction done with V_WMMA_F32_16X16X4_F32 (ones in B).
__global__ __launch_bounds__(256) void diag_offdiag_pass1(
    const float* __restrict__ x, float* __restrict__ ws) {
  const int tid = threadIdx.x;
  const v4f* p = (const v4f*)x + (size_t)blockIdx.x * 32768u + (unsigned)tid;

  v4f a0 = {0.f, 0.f, 0.f, 0.f}, a1 = a0, a2 = a0, a3 = a0;
  // 128 float4 loads per thread, batched 4-wide so 4 b128 NT loads are in flight.
  for (int k = 0; k < 128; k += 4) {
    v4f x0 = __builtin_nontemporal_load(p + (size_t)(k + 0) * 256);
    v4f x1 = __builtin_nontemporal_load(p + (size_t)(k + 1) * 256);
    v4f x2 = __builtin_nontemporal_load(p + (size_t)(k + 2) * 256);
    v4f x3 = __builtin_nontemporal_load(p + (size_t)(k + 3) * 256);
    a0 += x0; a1 += x1; a2 += x2; a3 += x3;
  }
  v4f acc = (a0 + a1) + (a2 + a3);

  __shared__ v4f lds[256];
  lds[tid] = acc;
  __syncthreads();

  // Wave 0 (lanes 0..31, full EXEC inside this branch) reduces the 256 thread
  // partials. Channel ch (0..63) is covered by the 16 threads t == (ch>>2) mod 16,
  // component ch&3. Pack 4 sources per step into the A matrix (16x4 f32 layout:
  // lane&15 = M, K = vgpr + 2*(lane>>4)), multiply by all-ones B:
  //   D[m,n] = C[m,n] + sum_k A[m,k].
  if (tid < 32) {
    const float* lf = (const float*)lds;
    const int m = tid & 15;   // row M = channel within 16-channel tile
    const int h = tid >> 4;   // half selects K = {2h, 2h+1} per A-VGPR
    const v2f bones = {1.0f, 1.0f};
#pragma unroll
    for (int g = 0; g < 4; ++g) {        // channel tiles 16g..16g+15
      const int tbase = 4 * g + (m >> 2);  // thread index base for this channel
      const int comp  = m & 3;             // float4 component for this channel
      v8f c = {0.f, 0.f, 0.f, 0.f, 0.f, 0.f, 0.f, 0.f};
#pragma unroll
      for (int r = 0; r < 4; ++r) {      // 4 accumulation steps cover sources 0..15
        v2f av;
        av.x = lf[(tbase + 16 * (4 * r + 0 + 2 * h)) * 4 + comp]; // K = 2h
        av.y = lf[(tbase + 16 * (4 * r + 1 + 2 * h)) * 4 + comp]; // K = 2h+1
        c = __builtin_amdgcn_wmma_f32_16x16x4_f32(
            /*neg_a=*/false, av, /*neg_b=*/false, bones,
            /*c_mod=*/(short)0, c, /*reuse_a=*/false, /*reuse_b=*/false);
      }
      // D rows: lane 0 holds M=0..7 in c[0..7]; lane 16 holds M=8..15.
      if (m == 0) {
        float* o = ws + (size_t)blockIdx.x * 64 + 16 * g + 8 * h;
#pragma unroll
        for (int j = 0; j < 8; ++j) o[j] = c[j];
      }
    }
  }
}

// Pass 2: one block per batch. Deterministically fold the 128 slab partials and
// the 512 diagonal rows, then write diag and (total - diag).
__global__ __launch_bounds__(256) void diag_offdiag_pass2(
    const float* __restrict__ x, const float* __restrict__ ws,
    float* __restrict__ out) {
  const int b = blockIdx.x;       // 0..15
  const int t = threadIdx.x;      // 0..255
  const int c = t & 63;
  const int q = t >> 6;           // 4 chunks

  // Diagonal: sum_i x[b,i,i,c] over this thread's i-chunk (coalesced over c).
  float dsum = 0.f;
  for (int i = q * 128; i < q * 128 + 128; ++i) {
    size_t idx = (((size_t)(b * 512 + i)) * 512 + (size_t)i) * 64 + (size_t)c;
    dsum += x[idx];
  }

  // Totals: fold the 128 per-slab partials belonging to this batch.
  float tsum = 0.f;
  for (int pIdx = q * 32; pIdx < q * 32 + 32; ++pIdx) {
    tsum += ws[((size_t)(b * 128 + pIdx)) * 64 + (size_t)c];
  }

  __shared__ float sd[4][64];
  __shared__ float st[4][64];
  sd[q][c] = dsum;
  st[q][c] = tsum;
  __syncthreads();

  if (t < 64) {
    float D = (sd[0][t] + sd[1][t]) + (sd[2][t] + sd[3][t]);
    float T = (st[0][t] + st[1][t]) + (st[2][t] + st[3][t]);
    out[b * 128 + t]       = D;      // diag_sum
    out[b * 128 + 64 + t]  = T - D;  // off-diag = total - diag
  }
}

extern "C" void kernel_launch(void* const* d_in, const int* in_sizes, int n_in,
                              void* d_out, int out_size, void* d_ws, size_t ws_size,
                              hipStream_t stream) {
  (void)in_sizes; (void)n_in; (void)out_size; (void)ws_size;
  const float* x = (const float*)d_in[0];
  float* out = (float*)d_out;
  float* ws  = (float*)d_ws;   // needs 2048 * 64 * 4 B = 512 KB

  diag_offdiag_pass1<<<2048, 256, 0, stream>>>(x, ws);
  diag_offdiag_pass2<<<16,   256, 0, stream>>>(x, ws, out);
}